// FDESampler_62431644614853
// MI455X (gfx1250) — compile-verified
//
#include <hip/hip_runtime.h>
#include <hip/hip_bf16.h>

#define Hh 488
#define KK 6
#define RECLEN 5
#define AW 484            // Hh - RECLEN + 1
#define NT 1024
#define THRF 12.0f        // 3.0 / 0.25
#define ITERS 10
#define RADIUS 2
#define TILE_MAX 26       // bounding box of d<12 disk is <= 26x26 integer pixels

typedef float v2f __attribute__((ext_vector_type(2)));
typedef float v8f __attribute__((ext_vector_type(8)));
typedef unsigned int u32x4 __attribute__((ext_vector_type(4)));
typedef int i32x4 __attribute__((ext_vector_type(4)));
typedef int i32x8 __attribute__((ext_vector_type(8)));

// ---- Tensor Data Mover: DMA a packed nr x nc float32 tile (row stride Hh) to LDS ----
// D# layout per CDNA5 ISA ch.8: group0 = {flags, lds_addr, global_addr, type=2},
// group1 = {data_size, tensor dims, tile dims, strides}. Groups 2/3 zero (2-D tensor).
__device__ __forceinline__ void tdm_load_tile_f32(const float* gsrc, unsigned lds_byte_addr,
                                                  int nr, int nc) {
  unsigned long long ga = (unsigned long long)(uintptr_t)gsrc;
  u32x4 g0;
  g0[0] = 1u;                                              // count=1, user descriptor
  g0[1] = lds_byte_addr;                                   // LDS destination (bytes)
  g0[2] = (unsigned)(ga & 0xFFFFFFFFu);                    // global_addr[31:0]
  g0[3] = (unsigned)((ga >> 32) & 0x01FFFFFFu) | 0x80000000u; // addr[56:32] | type=2
  i32x8 g1;
  g1[0] = (int)(2u << 16);                                 // data_size=2 -> 4 bytes
  g1[1] = (int)((unsigned)Hh << 16);                       // tensor_dim0[15:0]=488
  g1[2] = (int)((unsigned)Hh << 16);                       // td0 hi=0 | tensor_dim1 lo=488
  g1[3] = (int)((unsigned)nc << 16);                       // td1 hi=0 | tile_dim0=nc
  g1[4] = (int)((unsigned)nr & 0xFFFFu);                   // tile_dim1=nr, tile_dim2=0
  g1[5] = (int)Hh;                                         // tensor_dim0_stride[31:0]=488
  g1[6] = 0;                                               // stride0 hi | stride1 lo
  g1[7] = 0;                                               // tensor_dim1_stride hi
  i32x4 z4 = {0, 0, 0, 0};
#if __has_include(<hip/amd_detail/amd_gfx1250_TDM.h>)
  i32x8 z8 = {0, 0, 0, 0, 0, 0, 0, 0};
  __builtin_amdgcn_tensor_load_to_lds(g0, g1, z4, z4, z8, 0);   // clang-23 / therock form
#else
  __builtin_amdgcn_tensor_load_to_lds(g0, g1, z4, z4, 0);       // ROCm 7.2 form
#endif
}

// Heatmap value with the already-extracted 5x5 windows masked to zero
// (reference zeroes a heatmap copy; we mask instead so d_in stays untouched).
__device__ __forceinline__ float masked_val(const float* __restrict__ hm, int i, int j,
                                            const int* wr, const int* wc, int nwin) {
  float v = hm[i * Hh + j];
  for (int w = 0; w < nwin; ++w) {
    if (i >= wr[w] && i < wr[w] + RECLEN && j >= wc[w] && j < wc[w] + RECLEN) v = 0.f;
  }
  return v;
}

// ---------------- Phase 1: greedy 5x5-mean argmax peak extraction ----------------
__global__ void __launch_bounds__(NT) fde_peaks(const float* __restrict__ hm,
                                                float* __restrict__ cl_ws) {
  const int b = blockIdx.x;
  const int tid = threadIdx.x;
  const float* base = hm + (size_t)b * Hh * Hh;

  __shared__ float s_val[NT];
  __shared__ int   s_idx[NT];
  __shared__ int   wr[KK], wc[KK];
  __shared__ float s_max;

  for (int k = 0; k < KK; ++k) {
    // Sliding-window argmax of the 5x5 box sum (mean's /25 is monotonic -> skip).
    float best = -3.0e38f;
    int   bidx = 0x7fffffff;
    for (int c = tid; c < AW; c += NT) {
      __builtin_prefetch(&base[c], 0, 3);   // global_prefetch_b8
      float rs[RECLEN];
      float W = 0.f;
      for (int rr = 0; rr < RECLEN; ++rr) {
        float s = 0.f;
        for (int dc = 0; dc < RECLEN; ++dc) s += masked_val(base, rr, c + dc, wr, wc, k);
        rs[rr] = s; W += s;
      }
      if (W > best) { best = W; bidx = c; }          // flat = 0*AW + c
      int ring = 0;
      for (int r = 1; r < AW; ++r) {
        float s = 0.f;
        for (int dc = 0; dc < RECLEN; ++dc) s += masked_val(base, r + RECLEN - 1, c + dc, wr, wc, k);
        W += s - rs[ring];
        rs[ring] = s;
        ring = (ring + 1) % RECLEN;
        if (W > best) { best = W; bidx = r * AW + c; }  // ties keep smaller flat idx
      }
    }
    s_val[tid] = best; s_idx[tid] = bidx;
    for (int s = NT / 2; s > 0; s >>= 1) {
      __syncthreads();
      if (tid < s) {
        float ov = s_val[tid + s]; int oi = s_idx[tid + s];
        if (ov > s_val[tid] || (ov == s_val[tid] && oi < s_idx[tid])) {
          s_val[tid] = ov; s_idx[tid] = oi;
        }
      }
    }
    __syncthreads();
    const int flat = s_idx[0];
    const int r = flat / AW, c = flat % AW;

    // max value inside the winning 5x5 window
    if (tid == 0) {
      float mv = -3.0e38f;
      for (int dr = 0; dr < RECLEN; ++dr)
        for (int dc = 0; dc < RECLEN; ++dc)
          mv = fmaxf(mv, masked_val(base, r + dr, c + dc, wr, wc, k));
      s_max = mv;
    }
    __syncthreads();
    const float mv = s_max;

    // first flat index in the masked heatmap equal to that max (jnp.argmax(hm==mv))
    int mi = 0x7fffffff;
    for (int n = tid; n < Hh * Hh; n += NT)
      if (masked_val(base, n / Hh, n % Hh, wr, wc, k) == mv) { mi = n; break; }
    s_idx[tid] = mi;
    for (int s = NT / 2; s > 0; s >>= 1) {
      __syncthreads();
      if (tid < s) s_idx[tid] = min(s_idx[tid], s_idx[tid + s]);
    }
    __syncthreads();
    if (tid == 0) {
      const int fh = s_idx[0];
      cl_ws[((size_t)b * KK + k) * 2 + 0] = (float)(fh / Hh);
      cl_ws[((size_t)b * KK + k) * 2 + 1] = (float)(fh % Hh);
      wr[k] = r; wc[k] = c;          // zero this window for subsequent rounds
    }
    __syncthreads();
  }
}

// ------------- Phase 2+3: WMMA + TDM windowed mean-shift + confidences -------------
// One block per batch, 6 waves (wave32) = one wave per cluster.
// d^2[pixel, cluster] via V_WMMA_F32_16X16X4_F32:
//   A row p = (i, j, 1, i^2+j^2),  B col k = (-2ci, -2cj, ci^2+cj^2, 1)
// Each wave's bounding-box tile is DMA'd to LDS by the Tensor Data Mover.
__global__ void __launch_bounds__(192) fde_refine(const float* __restrict__ hm,
                                                  const float* __restrict__ cl_ws,
                                                  int* __restrict__ out_cl,
                                                  float* __restrict__ out_conf) {
  const int b    = blockIdx.x;
  const int tid  = threadIdx.x;
  const int kcl  = tid >> 5;        // wave id == cluster id
  const int lane = tid & 31;
  const int ncol = lane & 15;       // cluster column this lane owns in D
  const bool hi  = lane >= 16;
  const float* hmb = hm + (size_t)b * Hh * Hh;

  __shared__ float cx[KK], cy[KK], nx2[KK], ny2[KK];
  __shared__ float tiles[KK][TILE_MAX * TILE_MAX + 4];   // per-wave TDM destination

  if (tid < KK) {
    cx[tid] = cl_ws[((size_t)b * KK + tid) * 2 + 0];
    cy[tid] = cl_ws[((size_t)b * KK + tid) * 2 + 1];
  }
  __syncthreads();

  for (int it = 0; it < ITERS; ++it) {
    const float ci = cx[kcl], cj = cy[kcl];
    int r0 = max(0, (int)floorf(ci) - 12), r1 = min(Hh - 1, (int)ceilf(ci) + 12);
    int c0 = max(0, (int)floorf(cj) - 12), c1 = min(Hh - 1, (int)ceilf(cj) + 12);
    int nr = min(r1 - r0 + 1, TILE_MAX), nc = min(c1 - c0 + 1, TILE_MAX);
    int npix = (nr > 0 && nc > 0) ? nr * nc : 0;   // guards NaN clusters

    // Kick the TDM early: packed nr x nc tile -> this wave's LDS slot.
    if (npix > 0) {
      tdm_load_tile_f32(hmb + (size_t)r0 * Hh + c0,
                        (unsigned)(uintptr_t)(&tiles[kcl][0]), nr, nc);
    }

    // B operand while the DMA flies: per-cluster column; padded columns get a
    // huge |c|^2 so they never win the min and never pass the d<THR test.
    v2f Bop;
    if (!hi) {  // rows K=0,1
      if (ncol < KK) { Bop[0] = -2.f * cx[ncol]; Bop[1] = -2.f * cy[ncol]; }
      else           { Bop[0] = 0.f;             Bop[1] = 0.f; }
    } else {    // rows K=2,3
      if (ncol < KK) { Bop[0] = cx[ncol]*cx[ncol] + cy[ncol]*cy[ncol]; Bop[1] = 1.f; }
      else           { Bop[0] = 3.0e30f;                                Bop[1] = 1.f; }
    }

    __builtin_amdgcn_s_wait_tensorcnt(0);          // tile resident in LDS

    float sw = 0.f, swi = 0.f, swj = 0.f;
    for (int pb = 0; pb < npix; pb += 16) {        // uniform per wave -> EXEC all-1s
      const int pidx = pb + ncol;
      float pi, pj;
      if (pidx < npix) { pi = (float)(r0 + pidx / nc); pj = (float)(c0 + pidx % nc); }
      else             { pi = -65536.f; pj = -65536.f; }   // pushed out by d>=THR
      v2f Aop;
      if (!hi) { Aop[0] = pi;  Aop[1] = pj; }               // K=0,1
      else     { Aop[0] = 1.f; Aop[1] = pi * pi + pj * pj; }// K=2,3
      v8f Cz = {};
      v8f d2 = __builtin_amdgcn_wmma_f32_16x16x4_f32(false, Aop, false, Bop,
                                                     (short)0, Cz, false, false);
      float dl[8], mm[8];
      for (int v = 0; v < 8; ++v)                  // raw v_sqrt_f32: 1 ULP is plenty here
        dl[v] = fmaxf(__builtin_amdgcn_sqrtf(fmaxf(d2[v], 0.f)), 0.001f);
      // m[n] = min over the 16 cluster columns (lanes 0-15 / 16-31 hold disjoint pixels)
      for (int v = 0; v < 8; ++v) {
        float x = dl[v];
        for (int off = 8; off >= 1; off >>= 1) x = fminf(x, __shfl_xor(x, off, 16));
        mm[v] = x;
      }
      if (ncol == kcl) {   // only the owning cluster's column accumulates
        for (int v = 0; v < 8; ++v) {
          const int p2 = pb + v + (hi ? 8 : 0);   // D row M = v (+8 for upper lanes)
          if (p2 < npix) {
            const float d = dl[v];
            if (d < THRF) {
              const float h = tiles[kcl][p2];     // packed tile: p2 == row*nc+col
              const float w = (h * mm[v]) * __builtin_amdgcn_rcpf(d * d);
              const float fi = (float)(r0 + p2 / nc), fj = (float)(c0 + p2 % nc);
              sw += w; swi += w * fi; swj += w * fj;
            }
          }
        }
      }
    }
    for (int off = 16; off >= 1; off >>= 1) {
      sw  += __shfl_xor(sw,  off, 32);
      swi += __shfl_xor(swi, off, 32);
      swj += __shfl_xor(swj, off, 32);
    }
    if (lane == 0) { nx2[kcl] = swi / sw; ny2[kcl] = swj / sw; }
    __syncthreads();
    if (tid < KK) { cx[tid] = nx2[tid]; cy[tid] = ny2[tid]; }
    __syncthreads();
  }

  if (lane == 0) {
    const int ri = (int)rintf(cx[kcl]);     // jnp.round = half-to-even
    const int rj = (int)rintf(cy[kcl]);
    const int rs = min(max(ri - RADIUS, 0), Hh - 2 * RADIUS);  // dynamic_slice clamp
    const int cs = min(max(rj - RADIUS, 0), Hh - 2 * RADIUS);
    float conf = 0.f;
    for (int dr = 0; dr < 2 * RADIUS; ++dr)
      for (int dc = 0; dc < 2 * RADIUS; ++dc)
        conf += hmb[(rs + dr) * Hh + (cs + dc)];
    out_cl[((size_t)b * KK + kcl) * 2 + 0] = ri;
    out_cl[((size_t)b * KK + kcl) * 2 + 1] = rj;
    out_conf[(size_t)b * KK + kcl] = conf;
  }
}

extern "C" void kernel_launch(void* const* d_in, const int* in_sizes, int n_in,
                              void* d_out, int out_size, void* d_ws, size_t ws_size,
                              hipStream_t stream) {
  const float* hm = (const float*)d_in[0];
  const int B = in_sizes[0] / (Hh * Hh);            // 32
  float* cl_ws   = (float*)d_ws;                    // B*K*2 floats of scratch
  int*   out_cl  = (int*)d_out;                     // int32 clusters, raw bits first
  float* out_conf = (float*)d_out + (size_t)B * KK * 2;  // then float confidences

  fde_peaks<<<B, NT, 0, stream>>>(hm, cl_ws);
  fde_refine<<<B, 192, 0, stream>>>(hm, cl_ws, out_cl, out_conf);
}